// SelfAttention_41661182771778
// MI455X (gfx1250) — compile-verified
//
#include <hip/hip_runtime.h>

// Problem shape (from reference): B=16, M=2048, H=64, all float32.
// out[b] = (Q K^T) V = Q (K^T V)  -- softmax/span branch is dead code, so the
// O(M^2) attention matrix never needs to be formed: K^T V is just 64x64.
#define B_SZ 16
#define M_SZ 2048
#define H_SZ 64

typedef __attribute__((ext_vector_type(2))) float v2f;
typedef __attribute__((ext_vector_type(8))) float v8f;

// D(16x16,f32) += A(16x4,f32) x B(4x16,f32), wave32.
// Lane layout (CDNA5 ISA):
//   A: lane l (row = l%16) holds A[row, kb+v] in VGPR v, kb = 2*(l/16)
//   B: lane l (col = l%16) holds B[kb+v, col] in VGPR v
//   D: lane l holds D[r + 8*(l/16), l%16] in VGPR r (r=0..7)
static __device__ __forceinline__ v8f wmma_f32(v2f a, v2f b, v8f c) {
  return __builtin_amdgcn_wmma_f32_16x16x4_f32(
      /*neg_a=*/false, a, /*neg_b=*/false, b,
      /*c_mod=*/(short)0, c, /*reuse_a=*/false, /*reuse_b=*/false);
}

// ---------------------------------------------------------------------------
// Stage 1: S[b] = K[b]^T @ V[b]   (64x64 per batch)
// grid.x = B*16 (one workgroup per 16x16 tile of S), 512 threads = 16 waves.
// Wave w reduces M-slice [w*128, w*128+128) with K=4 WMMAs, then a fixed-order
// cross-wave LDS reduction (deterministic, no float atomics). K/V re-reads
// across the 4 tile rows/cols are served by the 192MB L2 (K+V = 16MB).
// ---------------------------------------------------------------------------
__global__ __launch_bounds__(512) void kvt_stage1(const float* __restrict__ K,
                                                  const float* __restrict__ V,
                                                  float* __restrict__ S) {
  const int b  = blockIdx.x >> 4;      // batch
  const int t  = blockIdx.x & 15;      // tile id in 4x4 grid of S
  const int ti = t >> 2;               // S row tile  -> H-slice of K
  const int tj = t & 3;                // S col tile  -> H-slice of V
  const int wave = threadIdx.x >> 5;
  const int lane = threadIdx.x & 31;
  const int row  = lane & 15;          // A row / B col within tile
  const int kb   = (lane >> 4) << 1;   // K-pair base: 0 or 2

  const float* Kb = K + (size_t)b * M_SZ * H_SZ;
  const float* Vb = V + (size_t)b * M_SZ * H_SZ;
  const int ka = ti * 16 + row;        // column of K (H index) this lane reads
  const int va = tj * 16 + row;        // column of V (H index) this lane reads

  v8f acc = {};
  const int m0 = wave * (M_SZ / 16);
  #pragma unroll 4
  for (int m = m0; m < m0 + (M_SZ / 16); m += 4) {
    v2f a, bb;
    // A[i,k] = K[m+k, ti*16+i]  (K^T tile)
    a.x  = Kb[(size_t)(m + kb + 0) * H_SZ + ka];
    a.y  = Kb[(size_t)(m + kb + 1) * H_SZ + ka];
    // B[k,j] = V[m+k, tj*16+j]
    bb.x = Vb[(size_t)(m + kb + 0) * H_SZ + va];
    bb.y = Vb[(size_t)(m + kb + 1) * H_SZ + va];
    acc = wmma_f32(a, bb, acc);
  }

  // Deterministic cross-wave reduction through LDS.
  __shared__ float red[16][256];       // 16 KB
  #pragma unroll
  for (int r = 0; r < 8; ++r) {
    const int e = (r + ((lane >> 4) << 3)) * 16 + (lane & 15); // rowD*16+colD
    red[wave][e] = acc[r];
  }
  __syncthreads();

  if (threadIdx.x < 256) {
    const int e = threadIdx.x;
    float s = 0.0f;
    #pragma unroll
    for (int w = 0; w < 16; ++w) s += red[w][e];
    const int sr = ti * 16 + (e >> 4);
    const int sc = tj * 16 + (e & 15);
    S[(size_t)b * H_SZ * H_SZ + sr * H_SZ + sc] = s;
  }
}

// ---------------------------------------------------------------------------
// Stage 2: out[b] = Q[b] @ S[b]    ([2048x64] x [64x64])
// grid.x = B * (M/128): each 256-thread block (8 waves) computes 128 output
// rows. Wave w owns the 16-row tile at m0+16w and ALL FOUR 16-col n-tiles
// (4 accumulators), so each A-fragment LDS read feeds 4 WMMAs and the S
// staging cost is amortized over 128 rows. LDS fills are b128-vectorized;
// row stride 68 floats (272B) keeps 16B alignment and makes the column-wise
// A reads bank-conflict-free (bank = (4*row + k) mod 64).
// ---------------------------------------------------------------------------
#define PAD_STRIDE 68
__global__ __launch_bounds__(256) void qs_stage2(const float* __restrict__ Q,
                                                 const float* __restrict__ S,
                                                 float* __restrict__ Out) {
  const int b  = blockIdx.x >> 4;            // batch
  const int m0 = (blockIdx.x & 15) << 7;     // block row base (128 rows)
  const int wave = threadIdx.x >> 5;
  const int lane = threadIdx.x & 31;
  const int row  = lane & 15;
  const int kb   = (lane >> 4) << 1;

  __shared__ float Qs[128 * PAD_STRIDE];     // 128x64 tile, padded (~34.8 KB)
  __shared__ float Ss[64 * PAD_STRIDE];      // 64x64, padded      (~17.4 KB)

  const float4* Qb4 = (const float4*)(Q + ((size_t)b * M_SZ + m0) * H_SZ);
  const float4* Sb4 = (const float4*)(S + (size_t)b * H_SZ * H_SZ);

  // Coalesced b128 cooperative fills (rows are 16 float4s).
  #pragma unroll
  for (int i = threadIdx.x; i < 128 * 16; i += 256) {
    const int r = i >> 4, c4 = i & 15;
    *(float4*)&Qs[r * PAD_STRIDE + c4 * 4] = Qb4[i];
  }
  #pragma unroll
  for (int i = threadIdx.x; i < 64 * 16; i += 256) {
    const int r = i >> 4, c4 = i & 15;
    *(float4*)&Ss[r * PAD_STRIDE + c4 * 4] = Sb4[i];
  }
  __syncthreads();

  v8f acc0 = {}, acc1 = {}, acc2 = {}, acc3 = {};
  const int qrow = (wave << 4) + row;        // this lane's A row in Qs
  #pragma unroll
  for (int k = 0; k < H_SZ; k += 4) {
    v2f a, b0, b1, b2, b3;
    a.x  = Qs[qrow * PAD_STRIDE + k + kb + 0];        // A[row, k+kb+v]
    a.y  = Qs[qrow * PAD_STRIDE + k + kb + 1];
    const int s0 = (k + kb + 0) * PAD_STRIDE + row;
    const int s1 = (k + kb + 1) * PAD_STRIDE + row;
    b0.x = Ss[s0 +  0]; b0.y = Ss[s1 +  0];           // B[k+kb+v, n0+col]
    b1.x = Ss[s0 + 16]; b1.y = Ss[s1 + 16];
    b2.x = Ss[s0 + 32]; b2.y = Ss[s1 + 32];
    b3.x = Ss[s0 + 48]; b3.y = Ss[s1 + 48];
    acc0 = wmma_f32(a, b0, acc0);
    acc1 = wmma_f32(a, b1, acc1);
    acc2 = wmma_f32(a, b2, acc2);
    acc3 = wmma_f32(a, b3, acc3);
  }

  // D[r + 8*(lane/16), lane%16] -> Out[b, m0+16w+rowD, 16j+colD];
  // per (r,j) the 16 lanes of a half-wave write one contiguous 64B run.
  float* Ob = Out + ((size_t)b * M_SZ + m0 + (wave << 4)) * H_SZ;
  #pragma unroll
  for (int r = 0; r < 8; ++r) {
    const int rowD = r + ((lane >> 4) << 3);
    float* p = Ob + (size_t)rowD * H_SZ + (lane & 15);
    p[ 0] = acc0[r];
    p[16] = acc1[r];
    p[32] = acc2[r];
    p[48] = acc3[r];
  }
}

extern "C" void kernel_launch(void* const* d_in, const int* in_sizes, int n_in,
                              void* d_out, int out_size, void* d_ws, size_t ws_size,
                              hipStream_t stream) {
  const float* Q = (const float*)d_in[0];
  const float* K = (const float*)d_in[1];
  // d_in[2] = span: dead code in the reference (output bypasses the
  // softmax/selection branch), so it is intentionally unused.
  const float* V = (const float*)d_in[3];
  float* Out = (float*)d_out;
  float* S   = (float*)d_ws;   // B*64*64 floats = 256 KB, fully overwritten

  kvt_stage1<<<B_SZ * 16, 512, 0, stream>>>(K, V, S);
  qs_stage2<<<B_SZ * (M_SZ / 128), 256, 0, stream>>>(Q, S, Out);
}